// ESPnetStatistic_8022998909740
// MI455X (gfx1250) — compile-verified
//
#include <hip/hip_runtime.h>
#include <hip/hip_bf16.h>

#define TPB    256          // threads per block (8 wave32)
#define DEPTH  4            // async pipeline depth (chunks in flight)
#define FTPB   256          // finalize block size
#define MAXBINS 1024
#define LOG2E  1.4426950408889634f

__device__ __forceinline__ float exp2fast(float x) {
    return __builtin_amdgcn_exp2f(x);   // v_exp_f32 (native base-2)
}

// Combine two online-softmax partials (m1,s1) <- (m1,s1) + (m2,s2)
__device__ __forceinline__ void combine(float& m, float& s, float m2, float s2) {
    float M = fmaxf(m, m2);
    s = s * exp2fast(m - M) + s2 * exp2fast(m2 - M);
    m = M;
}

// Issue a per-lane 16B async copy Global -> LDS (CDNA5 path, tracked by ASYNCcnt).
__device__ __forceinline__ void async_copy16(const float4* gsrc, float4* lds_dst) {
    unsigned      lds_addr = (unsigned)(uintptr_t)lds_dst;           // low 32b = LDS byte addr
    unsigned long long gad = (unsigned long long)(uintptr_t)gsrc;    // VGPR pair
    asm volatile("global_load_async_to_lds_b128 %0, %1, off th:TH_LOAD_NT"
                 :: "v"(lds_addr), "v"(gad) : "memory");
}

// One block per row: single-pass online softmax over V logits, write p (or -1) per row.
__global__ __launch_bounds__(TPB) void row_softmax_gather(
    const float* __restrict__ x, const int* __restrict__ ys,
    const int* __restrict__ ignore_id_p, float* __restrict__ pbuf, int V)
{
    __shared__ float4 stage[DEPTH * TPB];   // 16 KB staging, 4-deep double buffer
    __shared__ float  red_m[TPB / 32];
    __shared__ float  red_s[TPB / 32];

    const int row = blockIdx.x;
    const int tid = threadIdx.x;
    const float* rowp = x + (size_t)row * (size_t)V;
    const float4* rowq = (const float4*)rowp;

    const int nq      = V >> 2;                       // # of float4 in the row
    const int nchunks = (nq + TPB - 1) / TPB;

    const int tgt = ys[row];
    const int ign = *ignore_id_p;

    // ---- prologue: fill the async pipeline ----
    int pre = nchunks < DEPTH ? nchunks : DEPTH;
    for (int c = 0; c < pre; ++c) {
        int gi = c * TPB + tid; gi = gi < nq ? gi : nq - 1;           // clamp (uniform issue)
        async_copy16(rowq + gi, &stage[(c % DEPTH) * TPB + tid]);
    }

    // ---- streaming online softmax (base-2 domain) ----
    float m = -INFINITY, s = 0.0f;
    for (int c = 0; c < nchunks; ++c) {
        if (c + DEPTH < nchunks) asm volatile("s_wait_asynccnt 3" ::: "memory");
        else                     asm volatile("s_wait_asynccnt 0" ::: "memory");

        float4 v = stage[(c % DEPTH) * TPB + tid];    // ds_load_b128, own 16B only
        int gi = c * TPB + tid;
        if (gi < nq) {
            float y0 = v.x * LOG2E, y1 = v.y * LOG2E;
            float y2 = v.z * LOG2E, y3 = v.w * LOG2E;
            float mv = fmaxf(fmaxf(y0, y1), fmaxf(y2, y3));
            float M  = fmaxf(m, mv);
            s = s * exp2fast(m - M)
              + exp2fast(y0 - M) + exp2fast(y1 - M)
              + exp2fast(y2 - M) + exp2fast(y3 - M);
            m = M;
        }
        // LDS reads must land before this slot is overwritten by the next issue.
        asm volatile("s_wait_dscnt 0" ::: "memory");
        int nxt = c + DEPTH;
        if (nxt < nchunks) {
            int gi2 = nxt * TPB + tid; gi2 = gi2 < nq ? gi2 : nq - 1;
            async_copy16(rowq + gi2, &stage[(nxt % DEPTH) * TPB + tid]);
        }
    }

    // leftover scalars if V % 4 != 0
    int r = V & 3;
    if (tid < r) {
        float y = rowp[V - r + tid] * LOG2E;
        float M = fmaxf(m, y);
        s = s * exp2fast(m - M) + exp2fast(y - M);
        m = M;
    }

    // ---- wave32 reduction ----
    for (int off = 16; off > 0; off >>= 1) {
        float m2 = __shfl_xor(m, off, 32);
        float s2 = __shfl_xor(s, off, 32);
        combine(m, s, m2, s2);
    }
    if ((tid & 31) == 0) { red_m[tid >> 5] = m; red_s[tid >> 5] = s; }
    __syncthreads();

    if (tid == 0) {
        float M = red_m[0], S = red_s[0];
        for (int w = 1; w < TPB / 32; ++w) combine(M, S, red_m[w], red_s[w]);
        float p = -1.0f;                               // sentinel: padding position
        if (tgt != ign) {
            float yt = rowp[tgt] * LOG2E;
            p = exp2fast(yt - M) / S;
        }
        pbuf[row] = p;
    }
}

// Single block: deterministic mean + histogram (reproducing the reference's
// operator-precedence bug: counts[0] = #{0<p<1/bins}; counts[i>=1] = #{p > i/bins}).
__global__ __launch_bounds__(FTPB) void finalize_stats(
    const float* __restrict__ pbuf, int N,
    const int* __restrict__ bins_p, float* __restrict__ out)
{
    __shared__ unsigned hist[MAXBINS];
    __shared__ float    fsum[FTPB / 32];
    __shared__ unsigned vcnt[FTPB / 32];

    int bins = *bins_p; if (bins > MAXBINS) bins = MAXBINS; if (bins < 1) bins = 1;
    const int tid = threadIdx.x;

    for (int i = tid; i < bins; i += FTPB) hist[i] = 0u;
    __syncthreads();

    float   lsum = 0.0f;
    unsigned lcnt = 0u;
    const float fb = (float)bins;
    for (int i = tid; i < N; i += FTPB) {
        float p = pbuf[i];
        if (p >= 0.0f) {                // valid position
            lsum += p; lcnt++;
            if (p > 0.0f) {             // p==0 falls in no bin (strict > in reference)
                int j = (int)(p * fb);
                if (j > bins - 1) j = bins - 1;
                atomicAdd(&hist[j], 1u);
            }
        }
    }
    for (int off = 16; off > 0; off >>= 1) {
        lsum += __shfl_xor(lsum, off, 32);
        lcnt += __shfl_xor(lcnt, off, 32);
    }
    if ((tid & 31) == 0) { fsum[tid >> 5] = lsum; vcnt[tid >> 5] = lcnt; }
    __syncthreads();

    if (tid == 0) {
        float S = 0.0f; unsigned C = 0u;
        for (int w = 0; w < FTPB / 32; ++w) { S += fsum[w]; C += vcnt[w]; }
        out[0] = S / (float)C;          // confid_mean
    }
    __syncthreads();

    for (int i = tid; i < bins; i += FTPB) {
        unsigned c;
        if (i == 0) c = hist[0];                       // (0, 1/bins)
        else { c = 0u; for (int j = i; j < bins; ++j) c += hist[j]; }  // p > i/bins
        out[1 + i] = (float)c;
    }
}

extern "C" void kernel_launch(void* const* d_in, const int* in_sizes, int n_in,
                              void* d_out, int out_size, void* d_ws, size_t ws_size,
                              hipStream_t stream)
{
    const float* x      = (const float*)d_in[0];   // [B,T,V] f32 logits
    const int*   ys     = (const int*)  d_in[1];   // [B,T] targets
    const int*   bins_p = (const int*)  d_in[2];   // scalar
    const int*   ign_p  = (const int*)  d_in[3];   // scalar

    const int N = in_sizes[1];                     // B*T rows
    const int V = in_sizes[0] / N;                 // vocab

    float* pbuf = (float*)d_ws;                    // N floats, fully rewritten each call

    row_softmax_gather<<<N, TPB, 0, stream>>>(x, ys, ign_p, pbuf, V);
    finalize_stats<<<1, FTPB, 0, stream>>>(pbuf, N, bins_p, (float*)d_out);
}